// FusedOneMoE_35029753266180
// MI455X (gfx1250) — compile-verified
//
#include <hip/hip_runtime.h>
#include <hip/hip_bf16.h>

typedef __attribute__((ext_vector_type(2))) float v2f;
typedef __attribute__((ext_vector_type(8))) float v8f;
typedef __attribute__((ext_vector_type(4))) unsigned int u32x4;
typedef __attribute__((ext_vector_type(8))) int i32x8;
typedef __attribute__((ext_vector_type(4))) int i32x4;

#define B_SZ   4
#define S_SZ   2048
#define T_TOK  (B_SZ * S_SZ)   // 8192 tokens
#define E_DIM  1024
#define G_DIM  8
#define O_DIM  4096
#define TILE   128              // output tile edge
#define KCH    32               // K-chunk staged in LDS
#define LDB    (TILE + 1)       // 129 floats: B row stride (TDM pad 1 dword)
#define LDA3   (KCH + 2)        // 34 floats: k3 A row stride (TDM pad 2 dwords)
#define LDA2   (TILE + 1)       // 129 float2: k2 paired-A row stride

#if __has_builtin(__builtin_amdgcn_tensor_load_to_lds)
#define HAVE_TDM 1
#else
#define HAVE_TDM 0
#endif

// ---------------------------------------------------------------------------
// TDM 2D tile load: builds D# per CDNA5 ISA and issues
// tensor_load_to_lds (6-arg toolchain form: g0, g1, g2, g3, g4, cpol).
// data_size = 4 bytes. LDS row padding gives a bank-conflict-avoiding stride.
//   pad_int_enc: pad_interval encoding (row dwords = 2<<enc)
//   pad_amt_enc: pad_amount encoding (pad dwords = enc+1)
// ---------------------------------------------------------------------------
#if HAVE_TDM
__device__ __forceinline__ void tdm_load_2d(
    unsigned lds_off, const void* gptr,
    unsigned tensor_d0, unsigned tensor_d1,
    unsigned tile_d0, unsigned tile_d1,
    unsigned d0_stride,
    unsigned pad_int_enc, unsigned pad_amt_enc)
{
    unsigned long long ga = (unsigned long long)gptr;
    u32x4 g0;
    g0.x = 1u;                                   // count=1, user descriptor
    g0.y = lds_off;                              // lds_addr (bytes)
    g0.z = (unsigned)(ga & 0xFFFFFFFFu);         // global_addr[31:0]
    g0.w = (unsigned)((ga >> 32) & 0x01FFFFFFu)  // global_addr[56:32]
           | (2u << 30);                         // type = 2 ("image")
    i32x8 g1;
    unsigned dw0 = (2u << 16)                    // data_size = 4B
                 | (1u << 20)                    // pad_enable
                 | (pad_int_enc << 22)
                 | (pad_amt_enc << 25);
    g1[0] = (int)dw0;
    g1[1] = (int)((tensor_d0 & 0xFFFFu) << 16);                      // dim0 lo16
    g1[2] = (int)(((tensor_d0 >> 16) & 0xFFFFu)
                  | ((tensor_d1 & 0xFFFFu) << 16));                  // dim0 hi / dim1 lo
    g1[3] = (int)(((tensor_d1 >> 16) & 0xFFFFu)
                  | ((tile_d0 & 0xFFFFu) << 16));                    // dim1 hi / tile0
    g1[4] = (int)(tile_d1 & 0xFFFFu);                                // tile1, tile2=0
    g1[5] = (int)d0_stride;                                          // dim0_stride[31:0]
    g1[6] = 0;                                                       // stride hi, dim1_stride lo
    g1[7] = 0;
    i32x4 z4 = {0, 0, 0, 0};                     // groups 2/3 unused (2D tensor)
    i32x8 z8 = {0, 0, 0, 0, 0, 0, 0, 0};         // group 4 unused
    __builtin_amdgcn_tensor_load_to_lds(g0, g1, z4, z4, z8, 0);
}
#endif

__device__ __forceinline__ unsigned lds_offset_of(const void* p) {
    // Flat LDS addresses carry the aperture in the high 32 bits; low 32 bits
    // are the byte offset within the workgroup's LDS allocation.
    return (unsigned)(unsigned long long)p;
}

// ---------------------------------------------------------------------------
// Kernel 1: gate logits + top-2 + sigmoid weights (non-selected -> sigmoid(0)=0.5)
// One wave32 per token.
// ---------------------------------------------------------------------------
__global__ __launch_bounds__(256) void moe_gate_kernel(
    const float* __restrict__ hs, const float* __restrict__ WG,
    float* __restrict__ wOut)
{
    const int lane = threadIdx.x & 31;
    const int wid  = threadIdx.x >> 5;
    const int t    = blockIdx.x * 8 + wid;
    const float* row = hs + (size_t)t * E_DIM;

    float acc[G_DIM];
#pragma unroll
    for (int g = 0; g < G_DIM; ++g) acc[g] = 0.0f;

    for (int e = lane; e < E_DIM; e += 32) {
        const float x = row[e];
        const float* wg = WG + (size_t)e * G_DIM;
#pragma unroll
        for (int g = 0; g < G_DIM; ++g) acc[g] += x * wg[g];
    }
#pragma unroll
    for (int off = 16; off > 0; off >>= 1) {
#pragma unroll
        for (int g = 0; g < G_DIM; ++g) acc[g] += __shfl_xor(acc[g], off, 32);
    }
    if (lane == 0) {
        int i1 = 0;
#pragma unroll
        for (int g = 1; g < G_DIM; ++g) if (acc[g] > acc[i1]) i1 = g;
        int i2 = (i1 == 0) ? 1 : 0;
#pragma unroll
        for (int g = 0; g < G_DIM; ++g)
            if (g != i1 && acc[g] > acc[i2]) i2 = g;
#pragma unroll
        for (int g = 0; g < G_DIM; ++g) {
            float gv = (g == i1 || g == i2) ? acc[g] : 0.0f;  // masked logit
            wOut[(size_t)t * G_DIM + g] = 1.0f / (1.0f + __expf(-gv));
        }
    }
}

// ---------------------------------------------------------------------------
// Kernel 2: y = sum_g (w[:,g] * hs) @ Wi[g];  h = relu(y)^2  -> ws
//  - A staged manually (needs per-token gate scale) in a K-paired float2
//    layout so each WMMA A-fragment is one aligned ds_load_b64.
//  - B tile staged by the Tensor Data Mover with LDS pad -> 129-float rows.
// ---------------------------------------------------------------------------
__global__ __launch_bounds__(256) void moe_expert_gemm_kernel(
    const float* __restrict__ hs, const float* __restrict__ Wi,
    const float* __restrict__ w, float* __restrict__ hOut)
{
    // As: pair p holds {A[2p][t], A[2p+1][t]} as float2, row stride LDA2 pairs.
    __shared__ __align__(16) float As[(KCH / 2) * LDA2 * 2];
    __shared__ __align__(16) float Bs[KCH * LDB];
    __shared__ float sW[TILE];

    const int tid  = threadIdx.x;
    const int lane = tid & 31;
    const int wid  = tid >> 5;
    const int half = lane >> 4;      // 0: lanes 0-15, 1: lanes 16-31
    const int ln16 = lane & 15;
    const int tb   = (wid >> 2) * 64;  // wave token offset within tile
    const int nb   = (wid & 3) * 32;   // wave O-col offset within tile

    const int row0 = blockIdx.y * TILE;
    const int col0 = blockIdx.x * TILE;

    const unsigned bsOff = lds_offset_of(&Bs[0]);
    const v2f* ap = (const v2f*)As;

    v8f acc[4][2] = {};

    for (int g = 0; g < G_DIM; ++g) {
        __syncthreads();
        if (tid < TILE) sW[tid] = w[(size_t)(row0 + tid) * G_DIM + g];
        const float* wiG = Wi + (size_t)g * E_DIM * O_DIM;

        for (int e0 = 0; e0 < E_DIM; e0 += KCH) {
            __syncthreads();
#if HAVE_TDM
            if (wid == 0) {
                // B tile: KCH rows x TILE cols of Wi[g], row stride O_DIM.
                // pad_interval enc 6 = 128 dwords, pad_amount enc 0 = 1 dword.
                tdm_load_2d(bsOff, wiG + (size_t)e0 * O_DIM + col0,
                            O_DIM, E_DIM, TILE, KCH, O_DIM, 6u, 0u);
            }
#else
#pragma unroll
            for (int i = 0; i < (KCH * TILE) / 256; ++i) {
                int idx = i * 256 + tid;
                int n = idx & (TILE - 1);
                int e = idx >> 7;
                Bs[e * LDB + n] = wiG[(size_t)(e0 + e) * O_DIM + col0 + n];
            }
#endif
            // Stage A (transposed, gate-scaled, K-paired). Store banks
            // (e + 2t) mod 64 -> conflict-free.
#pragma unroll
            for (int i = 0; i < (KCH * TILE) / 256; ++i) {
                int idx = i * 256 + tid;
                int e = idx & (KCH - 1);
                int t = idx >> 5;
                As[(e >> 1) * (LDA2 * 2) + t * 2 + (e & 1)] =
                    sW[t] * hs[(size_t)(row0 + t) * E_DIM + e0 + e];
            }
#if HAVE_TDM
            if (wid == 0) __builtin_amdgcn_s_wait_tensorcnt(0);
#endif
            __syncthreads();

#pragma unroll
            for (int k0 = 0; k0 < KCH; k0 += 4) {
                // A 16x4 f32 frag: lanes 0-15 K=0/1, lanes 16-31 K=2/3 -> one b64.
                v2f a[4], b[2];
#pragma unroll
                for (int mt = 0; mt < 4; ++mt)
                    a[mt] = ap[((k0 >> 1) + half) * LDA2 + tb + mt * 16 + ln16];
#pragma unroll
                for (int nt = 0; nt < 2; ++nt) {
                    b[nt].x = Bs[(k0 + 2 * half + 0) * LDB + nb + nt * 16 + ln16];
                    b[nt].y = Bs[(k0 + 2 * half + 1) * LDB + nb + nt * 16 + ln16];
                }
#pragma unroll
                for (int mt = 0; mt < 4; ++mt)
#pragma unroll
                    for (int nt = 0; nt < 2; ++nt)
                        acc[mt][nt] = __builtin_amdgcn_wmma_f32_16x16x4_f32(
                            false, a[mt], false, b[nt],
                            (short)0, acc[mt][nt], false, false);
            }
        }
    }

    // Epilogue: h = relu(y)^2.  D layout: vgpr r -> m = r + 8*half, col = lane&15.
#pragma unroll
    for (int mt = 0; mt < 4; ++mt)
#pragma unroll
        for (int nt = 0; nt < 2; ++nt)
#pragma unroll
            for (int r = 0; r < 8; ++r) {
                float v = acc[mt][nt][r];
                v = (v > 0.0f) ? v * v : 0.0f;
                int m = r + 8 * half;
                hOut[(size_t)(row0 + tb + mt * 16 + m) * O_DIM
                     + col0 + nb + nt * 16 + ln16] = v;
            }
}

// ---------------------------------------------------------------------------
// Kernel 3: out = h @ down   ([8192,4096] x [4096,1024], fp32 WMMA)
// Both operand tiles staged by the TDM. A tile kept row-major [t][k] with a
// 34-float row stride (TDM pad 2 dwords): the A fragment {A[m][k],A[m][k+1]}
// is one aligned, conflict-free ds_load_b64.
// ---------------------------------------------------------------------------
__global__ __launch_bounds__(256) void moe_down_gemm_kernel(
    const float* __restrict__ h, const float* __restrict__ down,
    float* __restrict__ out)
{
    __shared__ __align__(16) float As[TILE * LDA3];   // [t][k], stride 34
    __shared__ __align__(16) float Bs[KCH * LDB];     // [k][n], stride 129

    const int tid  = threadIdx.x;
    const int lane = tid & 31;
    const int wid  = tid >> 5;
    const int half = lane >> 4;
    const int ln16 = lane & 15;
    const int tb   = (wid >> 2) * 64;
    const int nb   = (wid & 3) * 32;

    const int row0 = blockIdx.y * TILE;
    const int col0 = blockIdx.x * TILE;

    const unsigned asOff = lds_offset_of(&As[0]);
    const unsigned bsOff = lds_offset_of(&Bs[0]);

    v8f acc[4][2] = {};

    for (int k0g = 0; k0g < O_DIM; k0g += KCH) {
        __syncthreads();
#if HAVE_TDM
        if (wid == 0) {
            // A tile: TILE rows x KCH cols of h, row stride O_DIM.
            // pad_interval enc 4 = 32 dwords, pad_amount enc 1 = 2 dwords -> stride 34.
            tdm_load_2d(asOff, h + (size_t)row0 * O_DIM + k0g,
                        O_DIM, T_TOK, KCH, TILE, O_DIM, 4u, 1u);
            // B tile: KCH rows x TILE cols of down, row stride E_DIM.
            tdm_load_2d(bsOff, down + (size_t)k0g * E_DIM + col0,
                        E_DIM, O_DIM, TILE, KCH, E_DIM, 6u, 0u);
            __builtin_amdgcn_s_wait_tensorcnt(0);
        }
#else
#pragma unroll
        for (int i = 0; i < (KCH * TILE) / 256; ++i) {
            int idx = i * 256 + tid;
            int e = idx & (KCH - 1);
            int t = idx >> 5;
            As[t * LDA3 + e] = h[(size_t)(row0 + t) * O_DIM + k0g + e];
        }
#pragma unroll
        for (int i = 0; i < (KCH * TILE) / 256; ++i) {
            int idx = i * 256 + tid;
            int n = idx & (TILE - 1);
            int e = idx >> 7;
            Bs[e * LDB + n] = down[(size_t)(k0g + e) * E_DIM + col0 + n];
        }
#endif
        __syncthreads();

#pragma unroll
        for (int k0 = 0; k0 < KCH; k0 += 4) {
            v2f a[4], b[2];
#pragma unroll
            for (int mt = 0; mt < 4; ++mt)
                a[mt] = *(const v2f*)&As[(tb + mt * 16 + ln16) * LDA3
                                         + k0 + 2 * half];
#pragma unroll
            for (int nt = 0; nt < 2; ++nt) {
                b[nt].x = Bs[(k0 + 2 * half + 0) * LDB + nb + nt * 16 + ln16];
                b[nt].y = Bs[(k0 + 2 * half + 1) * LDB + nb + nt * 16 + ln16];
            }
#pragma unroll
            for (int mt = 0; mt < 4; ++mt)
#pragma unroll
                for (int nt = 0; nt < 2; ++nt)
                    acc[mt][nt] = __builtin_amdgcn_wmma_f32_16x16x4_f32(
                        false, a[mt], false, b[nt],
                        (short)0, acc[mt][nt], false, false);
        }
    }

#pragma unroll
    for (int mt = 0; mt < 4; ++mt)
#pragma unroll
        for (int nt = 0; nt < 2; ++nt)
#pragma unroll
            for (int r = 0; r < 8; ++r) {
                int m = r + 8 * half;
                out[(size_t)(row0 + tb + mt * 16 + m) * E_DIM
                    + col0 + nb + nt * 16 + ln16] = acc[mt][nt][r];
            }
}

// ---------------------------------------------------------------------------
extern "C" void kernel_launch(void* const* d_in, const int* in_sizes, int n_in,
                              void* d_out, int out_size, void* d_ws, size_t ws_size,
                              hipStream_t stream) {
    const float* hs   = (const float*)d_in[0];   // [B,S,E] fp32
    const float* WG   = (const float*)d_in[1];   // [E,G]
    const float* Wi   = (const float*)d_in[2];   // [G,E,O]
    const float* down = (const float*)d_in[3];   // [O,E]
    // d_in[4] = topk scalar (hardcoded 2)

    float* wBuf = (float*)d_ws;                       // T_TOK * G_DIM  (256 KB)
    float* hBuf = wBuf + (size_t)T_TOK * G_DIM;       // T_TOK * O_DIM  (128 MB)
    float* out  = (float*)d_out;                      // [B,S,E] fp32

    moe_gate_kernel<<<T_TOK / 8, 256, 0, stream>>>(hs, WG, wBuf);
    moe_expert_gemm_kernel<<<dim3(O_DIM / TILE, T_TOK / TILE), 256, 0, stream>>>(
        hs, Wi, wBuf, hBuf);
    moe_down_gemm_kernel<<<dim3(E_DIM / TILE, T_TOK / TILE), 256, 0, stream>>>(
        hBuf, down, out);
}